// SparseRouter_41944650613263
// MI455X (gfx1250) — compile-verified
//
#include <hip/hip_runtime.h>
#include <hip/hip_bf16.h>

typedef float v2f __attribute__((ext_vector_type(2)));
typedef float v8f __attribute__((ext_vector_type(8)));

#define TOKENS   16384
#define DMODEL   4096
#define NEXPERT  64
#define TILE_M   128
#define KC       32
#define XS_STR   36   // x tile stride: conflict-free A-frag reads, 16B-aligned float4 stores
#define WT_STR   34   // transposed W stride: 8B-aligned float2 B-frags, ~conflict-free reads
#define LG_STR   65   // logits stride: (row + j) % 64 -> conflict-free per-row scans

__global__ __launch_bounds__(256) void moe_router_kernel(
    const float* __restrict__ x, const float* __restrict__ W,
    const float* __restrict__ bias, float* __restrict__ out)
{
    __shared__ float xs[TILE_M * XS_STR];     // 18 KB   x tile [128][32] (padded)
    __shared__ float wt[NEXPERT * WT_STR];    // 8.5 KB  W tile transposed [64 experts][32 k]
    __shared__ float lg[TILE_M * LG_STR];     // 33 KB   logits [128][64] (padded)

    const int tid  = threadIdx.x;
    const int wave = tid >> 5;
    const int lane = tid & 31;
    const int hi   = lane >> 4;   // half-wave: K pair select / row+8
    const int l16  = lane & 15;
    const int row0 = blockIdx.x * TILE_M;

    // ---- staging assignments ----
    const int xrow  = tid >> 1;          // 0..127
    const int xcol0 = (tid & 1) * 16;    // 16 floats per thread of one token row
    const int wk    = tid >> 3;          // W row k (0..31) within chunk
    const int we0   = (tid & 7) * 8;     // 8 experts per thread

    const float* xbase = x + (size_t)(row0 + xrow) * DMODEL + xcol0;
    const float* wbase = W + (size_t)wk * NEXPERT + we0;

    // ---- software pipeline: prefetch chunk 0 into registers ----
    float4 xr0 = *(const float4*)(xbase +  0);
    float4 xr1 = *(const float4*)(xbase +  4);
    float4 xr2 = *(const float4*)(xbase +  8);
    float4 xr3 = *(const float4*)(xbase + 12);
    float4 wr0 = *(const float4*)(wbase +  0);
    float4 wr1 = *(const float4*)(wbase +  4);

    v8f acc[4] = {};   // 16 rows x 64 experts per wave

    for (int k0 = 0; k0 < DMODEL; k0 += KC) {
        // ---- drain staged registers into LDS ----
        float* xd = xs + xrow * XS_STR + xcol0;
        *(float4*)(xd +  0) = xr0;
        *(float4*)(xd +  4) = xr1;
        *(float4*)(xd +  8) = xr2;
        *(float4*)(xd + 12) = xr3;

        float* wd = wt + we0 * WT_STR + wk;       // transpose: wt[e][k], member access only
        wd[0 * WT_STR] = wr0.x;
        wd[1 * WT_STR] = wr0.y;
        wd[2 * WT_STR] = wr0.z;
        wd[3 * WT_STR] = wr0.w;
        wd[4 * WT_STR] = wr1.x;
        wd[5 * WT_STR] = wr1.y;
        wd[6 * WT_STR] = wr1.z;
        wd[7 * WT_STR] = wr1.w;

        __syncthreads();

        // ---- issue next chunk's global loads; overlap with WMMA below ----
        if (k0 + KC < DMODEL) {
            const float* xn = xbase + k0 + KC;
            xr0 = *(const float4*)(xn +  0);
            xr1 = *(const float4*)(xn +  4);
            xr2 = *(const float4*)(xn +  8);
            xr3 = *(const float4*)(xn + 12);
            const float* wn = wbase + (size_t)(k0 + KC) * NEXPERT;
            wr0 = *(const float4*)(wn + 0);
            wr1 = *(const float4*)(wn + 4);
        }

        // ---- 8 K-steps x 4 expert tiles of V_WMMA_F32_16X16X4_F32 ----
        const float* arow = xs + (wave * 16 + l16) * XS_STR + 2 * hi;
#pragma unroll
        for (int ks = 0; ks < KC / 4; ++ks) {
            v2f a = *(const v2f*)(arow + ks * 4);        // lanes0-15: K=0,1; lanes16-31: K=2,3
#pragma unroll
            for (int n = 0; n < 4; ++n) {
                v2f b = *(const v2f*)(wt + (n * 16 + l16) * WT_STR + ks * 4 + 2 * hi);
                acc[n] = __builtin_amdgcn_wmma_f32_16x16x4_f32(
                    false, a, false, b, (short)0, acc[n], false, false);
            }
        }
        __syncthreads();
    }

    // ---- spill logits: C/D layout VGPR v -> row v + 8*hi, col = lane&15 ----
#pragma unroll
    for (int n = 0; n < 4; ++n)
#pragma unroll
        for (int v = 0; v < 8; ++v)
            lg[(wave * 16 + v + 8 * hi) * LG_STR + n * 16 + l16] = acc[n][v];
    __syncthreads();

    // ---- softmax + top-2 + renormalize: one thread per token row ----
    if (tid < TILE_M) {
        const int gr = row0 + tid;
        float* rowlg = lg + tid * LG_STR;

        float mx = -3.402823466e38f;
#pragma unroll 8
        for (int j = 0; j < NEXPERT; ++j) {
            float l = rowlg[j] + bias[j];
            rowlg[j] = l;
            mx = fmaxf(mx, l);
        }

        float sum = 0.f;
        float e0 = -1.f, e1 = -1.f;
        int   i0 = 0,   i1 = 0;
#pragma unroll 8
        for (int j = 0; j < NEXPERT; ++j) {
            float e = expf(rowlg[j] - mx);
            rowlg[j] = e;
            sum += e;
            if (e > e0)      { e1 = e0; i1 = i0; e0 = e; i0 = j; }  // strict '>' = lowest-index tie-break
            else if (e > e1) { e1 = e;  i1 = j; }
        }
        const float inv = 1.f / sum;

        float* probs_out = out + (size_t)TOKENS * 4;      // after topk_p (2N) + topk_i (2N)
#pragma unroll 8
        for (int j = 0; j < NEXPERT; ++j)
            probs_out[(size_t)gr * NEXPERT + j] = rowlg[j] * inv;

        const float denom = e0 + e1;                      // renormalized top-2
        out[2 * gr + 0] = e0 / denom;
        out[2 * gr + 1] = e1 / denom;

        float* idx_out = out + (size_t)TOKENS * 2;        // indices, value-cast to float
        idx_out[2 * gr + 0] = (float)i0;
        idx_out[2 * gr + 1] = (float)i1;
    }
}

extern "C" void kernel_launch(void* const* d_in, const int* in_sizes, int n_in,
                              void* d_out, int out_size, void* d_ws, size_t ws_size,
                              hipStream_t stream) {
    const float* x  = (const float*)d_in[0];
    const float* W  = (const float*)d_in[1];
    const float* b  = (const float*)d_in[2];
    float*       out = (float*)d_out;

    dim3 grid(TOKENS / TILE_M);   // 128 blocks
    dim3 block(256);              // 8 wave32s
    moe_router_kernel<<<grid, block, 0, stream>>>(x, W, b, out);
}